// DeltaNetLayer_22488448762151
// MI455X (gfx1250) — compile-verified
//
#include <hip/hip_runtime.h>

// Gated delta-rule recurrence for MI455X (gfx1250, wave32).
//   M_t = M_{t-1} * clip(f⊗f, 0.8, 1) + (k*g)⊗(v*g) ;  o_t = q_t @ M_t
// grid = (B*H, NSPLIT); each 128-thread block owns 64 columns of one (b,h)
// state. State M^T lives in VGPRs in the exact A-operand layout of
// v_wmma_f32_16x16x4_f32: wave w, lane L -> column j = 16*(4*s+w)+(L&15),
// VGPR pair c -> rows i = 4c + {0,1} + 2*(L>=16). The per-step GEMV is 32
// chained WMMAs (2 accumulator chains); the elementwise gate update runs on
// the VALU/pk pipes and co-executes with the XDL pipe. Per-step q/f vectors
// stream global->LDS via async-to-LDS; chunk operand loads are contiguous
// ds_load_b64 software-pipelined one chunk ahead.

#define B_ 4
#define T_ 2048
#define H_ 16
#define D_ 128
#define F_MIN 0.8f
#define NSPLIT 2

typedef float v2f __attribute__((ext_vector_type(2)));
typedef float v8f __attribute__((ext_vector_type(8)));

#if __has_builtin(__builtin_amdgcn_global_load_async_to_lds_b128) && \
    __has_builtin(__builtin_amdgcn_s_wait_asynccnt)
#define USE_ASYNC 1
// Builtin signature (from hipcc diagnostic): param0 = int4-vector ptr in
// address_space(1) (prints as "__device__"), param1 = same in AS(3), then
// two immediate ints (offset, cpol).
typedef int vi4 __attribute__((vector_size(16)));
typedef __attribute__((address_space(1))) vi4 as1_vi4;
typedef __attribute__((address_space(3))) vi4 as3_vi4;
__device__ __forceinline__ void async_ld16(const float* gsrc, float* ldst) {
  __builtin_amdgcn_global_load_async_to_lds_b128(
      (as1_vi4*)gsrc, (as3_vi4*)ldst, 0, 0);
}
#else
#define USE_ASYNC 0
#endif

__global__ __launch_bounds__(128) void deltanet_gated_kernel(
    const float* __restrict__ qg, const float* __restrict__ kg_in,
    const float* __restrict__ vg_in, const float* __restrict__ fg,
    const float* __restrict__ gg, float* __restrict__ out)
{
  // Double-buffered per-step vectors + output staging (~4.5 KB LDS).
  __shared__ __align__(16) float FS[2][D_];   // f_t
  __shared__ __align__(16) float KS[2][D_];   // k_t * g_t
  __shared__ __align__(16) float VS[2][D_];   // v_t * g_t
  __shared__ __align__(16) float QS[2][D_];   // q_t
  __shared__ __align__(16) float OS[2][64];   // o_t staging (this block's cols)

  const int tid  = threadIdx.x;     // 0..127
  const int lane = tid & 31;        // wave32
  const int wav  = tid >> 5;        // 0..3
  const int hlf  = lane >> 4;
  const int jl   = lane & 15;
  const int s    = blockIdx.y;                  // column split 0..NSPLIT-1
  const int j    = (s * 4 + wav) * 16 + jl;     // column owned by this lane

  const int bh   = blockIdx.x;
  const int b    = bh / H_;
  const int head = bh % H_;
  const int base0   = (b * T_ * H_ + head) * D_;
  const int tstride = H_ * D_;

  // State M^T in WMMA-A layout: Mst[c] = rows i = 4c + {0,1} + 2*hlf, col j.
  v2f Mst[32];
#pragma unroll
  for (int c = 0; c < 32; ++c) { Mst[c].x = 0.0f; Mst[c].y = 0.0f; }

  // ---- prologue: stage step 0 ----
  float rkg, rvg;
  {
    int a = base0 + tid;
    float kk = kg_in[a], vv = vg_in[a], g2 = gg[a];
    rkg = kk * g2; rvg = vv * g2;
  }
#if USE_ASYNC
  if (wav == 0) {
    int a = base0 + 4 * lane;                    // 32 lanes x 16B = 512B
    async_ld16(qg + a, &QS[0][4 * lane]);
    async_ld16(fg + a, &FS[0][4 * lane]);
  }
#else
  float rq, rf;
  { int a = base0 + tid; rq = qg[a]; rf = fg[a]; }
#endif

  for (int t = 0; t < T_; ++t) {
    const int pb = t & 1;

    KS[pb][tid] = rkg; VS[pb][tid] = rvg;
#if USE_ASYNC
    __builtin_amdgcn_s_wait_asynccnt(0);  // q/f for step t landed in LDS
#else
    QS[pb][tid] = rq; FS[pb][tid] = rf;
#endif
    __syncthreads();

    // Prefetch t+1 (latency hidden under compute below); hint t+2 to L2.
    {
      int tn = (t + 1 < T_) ? t + 1 : t;
      int a  = base0 + tn * tstride + tid;
      float kk = kg_in[a], vv = vg_in[a], g2 = gg[a];
      rkg = kk * g2; rvg = vv * g2;
#if USE_ASYNC
      if (wav == 0) {
        int a4 = base0 + tn * tstride + 4 * lane;
        async_ld16(qg + a4, &QS[pb ^ 1][4 * lane]);
        async_ld16(fg + a4, &FS[pb ^ 1][4 * lane]);
      }
#else
      rq = qg[a]; rf = fg[a];
#endif
      int tp = (t + 2 < T_) ? t + 2 : T_ - 1;
      int ap = base0 + tp * tstride + tid;
      __builtin_prefetch(kg_in + ap, 0, 1);
      __builtin_prefetch(vg_in + ap, 0, 1);
      __builtin_prefetch(gg + ap, 0, 1);
    }

    const float fj  = FS[pb][j];   // column gate for this lane
    const float vgj = VS[pb][j];   // column of the rank-1 update

    // Contiguous operand pairs {x[4c+2h], x[4c+2h+1]}; pipeline 1 chunk ahead.
    const float* Fp = &FS[pb][2 * hlf];
    const float* Kp = &KS[pb][2 * hlf];
    const float* Qp = &QS[pb][2 * hlf];
    v2f f2 = *(const v2f*)Fp;
    v2f k2 = *(const v2f*)Kp;
    v2f q2 = *(const v2f*)Qp;

    v8f acc0 = {}; v8f acc1 = {};  // two chains to relax D->C RAW latency
#pragma unroll
    for (int c = 0; c < 32; ++c) {
      const int cn = ((c + 1) & 31) * 4;         // next chunk (c=31 wraps, harmless)
      const v2f fn = *(const v2f*)(Fp + cn);
      const v2f kn = *(const v2f*)(Kp + cn);
      const v2f qn = *(const v2f*)(Qp + cn);

      // Elementwise gated update: M = M*clip(fi*fj, .8, 1) + kg_i*vg_j
      v2f gt;
      gt.x = fminf(fmaxf(f2.x * fj, F_MIN), 1.0f);
      gt.y = fminf(fmaxf(f2.y * fj, F_MIN), 1.0f);
      v2f m = Mst[c];
      m.x = m.x * gt.x + k2.x * vgj;
      m.y = m.y * gt.y + k2.y * vgj;
      Mst[c] = m;

      // o[j] += M^T[j, 4c..4c+3] . q[4c..4c+3]  on the XDL pipe
      if (c & 1)
        acc1 = __builtin_amdgcn_wmma_f32_16x16x4_f32(
            false, m, false, q2, (short)0, acc1, false, false);
      else
        acc0 = __builtin_amdgcn_wmma_f32_16x16x4_f32(
            false, m, false, q2, (short)0, acc0, false, false);

      f2 = fn; k2 = kn; q2 = qn;
    }
    const v8f od = acc0 + acc1;

    // D layout: VGPR r, lane L -> row (r + 8*hlf), col replicated over N.
    if (jl == 0) {
#pragma unroll
      for (int r = 0; r < 8; ++r)
        OS[pb][wav * 16 + hlf * 8 + r] = od[r];
    }
    __syncthreads();

    if (tid < 64)
      out[base0 + t * tstride + s * 64 + tid] = OS[pb][tid];
  }
}

extern "C" void kernel_launch(void* const* d_in, const int* in_sizes, int n_in,
                              void* d_out, int out_size, void* d_ws, size_t ws_size,
                              hipStream_t stream) {
  const float* q = (const float*)d_in[0];
  const float* k = (const float*)d_in[1];
  const float* v = (const float*)d_in[2];
  const float* f = (const float*)d_in[3];
  const float* g = (const float*)d_in[4];
  float* out = (float*)d_out;

  dim3 grid(B_ * H_, NSPLIT);  // 128 blocks: 64 recurrences x 2 column halves
  dim3 block(128);             // 4 wave32 waves
  hipLaunchKernelGGL(deltanet_gated_kernel, grid, block, 0, stream,
                     q, k, v, f, g, out);
}